// DEDICOMDecoder_30477087932956
// MI455X (gfx1250) — compile-verified
//
#include <hip/hip_runtime.h>
#include <hip/hip_bf16.h>

typedef __attribute__((ext_vector_type(2))) float v2f;
typedef __attribute__((ext_vector_type(8))) float v8f;

#define EMB 128

// ---------------------------------------------------------------------------
// Phase 1: u[r, n] = sum_k z[r, k] * (d_r[k] * R[k, n])
// Block: 256 threads = 8 waves. Block handles 16 rows; wave w handles cols
// [16w, 16w+16). K marched 4 at a time with V_WMMA_F32_16X16X4_F32.
//
// A (16x4 f32) lane layout: lane L holds row M = L%16; half = L/16;
//   vgpr v holds K = k0 + 2*half + v          -> contiguous float2 from LDS.
// B (4x16 f32) lane layout: lane L holds col N = L%16; half = L/16;
//   vgpr v holds K = k0 + 2*half + v.
// C/D (16x16 f32): vgpr v, lane L -> M = v + 8*(L/16), N = L%16.
// ---------------------------------------------------------------------------
__global__ __launch_bounds__(256)
void dedicom_u_gemm_kernel(const float* __restrict__ z,
                           const float* __restrict__ R,
                           const float* __restrict__ D,
                           const int*   __restrict__ rel,
                           float* __restrict__ u,
                           int n_drugs) {
    __shared__ float lds_z[16 * EMB];   // 8 KB tile of z rows
    __shared__ float lds_dr[EMB];       // relation diagonal

    const int tid  = threadIdx.x;
    const int lane = tid & 31;
    const int wid  = tid >> 5;          // 0..7 -> N tile
    const int r0   = blockIdx.x * 16;
    const int relv = rel[0];

    // Stage 16x128 z tile (rows are contiguous -> one 8KB contiguous region).
    {
        const size_t base = (size_t)r0 * EMB;
        const size_t lim  = (size_t)n_drugs * EMB;
        size_t o0 = base + (size_t)tid * 4;
        size_t o1 = base + 4096 + (size_t)tid * 4;   // 4096 floats later
        float4 a = make_float4(0.f, 0.f, 0.f, 0.f);
        float4 b = make_float4(0.f, 0.f, 0.f, 0.f);
        if (o0 + 3 < lim) a = *(const float4*)(z + o0);
        if (o1 + 3 < lim) b = *(const float4*)(z + o1);
        *(float4*)&lds_z[tid * 4]        = a;
        *(float4*)&lds_z[4096 + tid * 4] = b;
        if (tid < 32) {
            *(float4*)&lds_dr[tid * 4] =
                *(const float4*)(D + (size_t)relv * EMB + tid * 4);
        }
    }
    __syncthreads();

    const int M    = lane & 15;     // A row (also B/D column index)
    const int half = lane >> 4;
    const int n0   = wid * 16;
    const int n    = n0 + M;        // B/D column for this lane

    v8f acc = {};

#pragma unroll 4
    for (int k0 = 0; k0 < EMB; k0 += 4) {
        const int kb = k0 + 2 * half;
        // A: two consecutive K values of this lane's z row (from LDS)
        float2 az = *(const float2*)&lds_z[M * EMB + kb];
        float2 dr = *(const float2*)&lds_dr[kb];
        v2f a; a.x = az.x; a.y = az.y;
        // B: fold d_r into R at load time (R is tiny, L2/L0 resident)
        v2f b;
        b.x = dr.x * R[(size_t)kb * EMB + n];
        b.y = dr.y * R[(size_t)(kb + 1) * EMB + n];
        acc = __builtin_amdgcn_wmma_f32_16x16x4_f32(
            false, a, false, b, (short)0, acc, false, false);
    }

    // Write D-matrix: vgpr v -> row r0 + v + 8*half, col n.
    // Fast path (tile fully in range, always true when n_drugs % 16 == 0):
    // straight-line stores with immediate row offsets, no exec-mask churn.
    float* ubase = u + ((size_t)(r0 + 8 * half)) * EMB + n;
    if (r0 + 15 < n_drugs) {
#pragma unroll
        for (int v = 0; v < 8; ++v) {
            ubase[(size_t)v * EMB] = acc[v];
        }
    } else {
#pragma unroll
        for (int v = 0; v < 8; ++v) {
            if (r0 + v + 8 * half < n_drugs) ubase[(size_t)v * EMB] = acc[v];
        }
    }
}

// ---------------------------------------------------------------------------
// Phase 2: one wave per edge: score = sum_e u[i][e] * z[j][e] * d_r[e]
// Each lane owns 4 contiguous columns -> float4 (b128) gathers, wave32
// shuffle reduction. Both gather tables fit in the 192MB L2.
// ---------------------------------------------------------------------------
__global__ __launch_bounds__(256)
void dedicom_edge_kernel(const float* __restrict__ u,
                         const float* __restrict__ z,
                         const float* __restrict__ D,
                         const int*   __restrict__ rel,
                         const int*   __restrict__ edge,
                         float* __restrict__ out,
                         int n_edges) {
    const int lane = threadIdx.x & 31;
    const int wid  = threadIdx.x >> 5;
    const long long e = (long long)blockIdx.x * 8 + wid;
    if (e >= n_edges) return;

    const int relv = rel[0];
    const int i = edge[e];
    const int j = edge[(size_t)n_edges + e];

    float4 uv = *(const float4*)(u + (size_t)i * EMB + lane * 4);
    float4 zv = *(const float4*)(z + (size_t)j * EMB + lane * 4);
    float4 dv = *(const float4*)(D + (size_t)relv * EMB + lane * 4);

    float s = uv.x * zv.x * dv.x + uv.y * zv.y * dv.y +
              uv.z * zv.z * dv.z + uv.w * zv.w * dv.w;

#pragma unroll
    for (int off = 16; off > 0; off >>= 1)
        s += __shfl_xor(s, off, 32);

    if (lane == 0) out[e] = s;
}

// ---------------------------------------------------------------------------
// Fallback (only if workspace is too small): fused wave-per-edge bilinear.
// ---------------------------------------------------------------------------
__global__ __launch_bounds__(256)
void dedicom_edge_full_kernel(const float* __restrict__ z,
                              const float* __restrict__ R,
                              const float* __restrict__ D,
                              const int*   __restrict__ rel,
                              const int*   __restrict__ edge,
                              float* __restrict__ out,
                              int n_edges) {
    __shared__ float zid[8][EMB];
    __shared__ float zjd[8][EMB];
    const int lane = threadIdx.x & 31;
    const int wid  = threadIdx.x >> 5;
    const long long e = (long long)blockIdx.x * 8 + wid;
    const bool active = (e < n_edges);
    const int relv = rel[0];

    if (active) {
        const int i = edge[e];
        const int j = edge[(size_t)n_edges + e];
        float4 dv = *(const float4*)(D + (size_t)relv * EMB + lane * 4);
        float4 zi = *(const float4*)(z + (size_t)i * EMB + lane * 4);
        float4 zj = *(const float4*)(z + (size_t)j * EMB + lane * 4);
        *(float4*)&zid[wid][lane * 4] =
            make_float4(zi.x * dv.x, zi.y * dv.y, zi.z * dv.z, zi.w * dv.w);
        *(float4*)&zjd[wid][lane * 4] =
            make_float4(zj.x * dv.x, zj.y * dv.y, zj.z * dv.z, zj.w * dv.w);
    }
    __syncthreads();
    if (!active) return;

    const int n = lane * 4;
    float a0 = 0.f, a1 = 0.f, a2 = 0.f, a3 = 0.f;
    for (int d = 0; d < EMB; ++d) {
        float a = zid[wid][d];
        const float* Rr = R + (size_t)d * EMB + n;
        a0 += a * Rr[0]; a1 += a * Rr[1]; a2 += a * Rr[2]; a3 += a * Rr[3];
    }
    float s = a0 * zjd[wid][n] + a1 * zjd[wid][n + 1] +
              a2 * zjd[wid][n + 2] + a3 * zjd[wid][n + 3];

#pragma unroll
    for (int off = 16; off > 0; off >>= 1)
        s += __shfl_xor(s, off, 32);

    if (lane == 0) out[e] = s;
}

extern "C" void kernel_launch(void* const* d_in, const int* in_sizes, int n_in,
                              void* d_out, int out_size, void* d_ws, size_t ws_size,
                              hipStream_t stream) {
    const float* z    = (const float*)d_in[0];   // (N, 128)
    const float* R    = (const float*)d_in[1];   // (128, 128)
    const float* D    = (const float*)d_in[2];   // (n_rel, 128)
    const int*   edge = (const int*)d_in[3];     // (2, B)
    const int*   rel  = (const int*)d_in[4];     // scalar
    float* out = (float*)d_out;

    const int n_drugs = in_sizes[0] / EMB;       // 50000
    const int n_edges = in_sizes[3] / 2;         // 1,000,000

    const size_t need = (size_t)n_drugs * EMB * sizeof(float);  // 25.6 MB
    const int edge_blocks = (n_edges + 7) / 8;

    if (ws_size >= need) {
        float* u = (float*)d_ws;
        const int m_blocks = (n_drugs + 15) / 16;
        dedicom_u_gemm_kernel<<<m_blocks, 256, 0, stream>>>(z, R, D, rel, u, n_drugs);
        dedicom_edge_kernel<<<edge_blocks, 256, 0, stream>>>(u, z, D, rel, edge, out, n_edges);
    } else {
        dedicom_edge_full_kernel<<<edge_blocks, 256, 0, stream>>>(z, R, D, rel, edge, out, n_edges);
    }
}